// InvariantMessagePassingTP_13572096655503
// MI455X (gfx1250) — compile-verified
//
#include <hip/hip_runtime.h>
#include <cstdint>

typedef __attribute__((ext_vector_type(2))) float v2f;
typedef __attribute__((ext_vector_type(8))) float v8f;

// ---------------------------------------------------------------------------
// Zero-fill output (harness poisons d_out; we accumulate with atomics).
// ---------------------------------------------------------------------------
__global__ void imp_zero_kernel(float* __restrict__ out, int n) {
  int i = blockIdx.x * blockDim.x + threadIdx.x;
  int i4 = i * 4;
  if (i4 + 3 < n) {
    float4 z = make_float4(0.f, 0.f, 0.f, 0.f);
    *(float4*)(out + i4) = z;
  } else {
    for (int k = i4; k < n; ++k) out[k] = 0.f;
  }
}

// ---------------------------------------------------------------------------
// One wave32 per edge. Per-edge compute cast as 16x4 @ 4x96 matmul:
//   A[lm, l] = edge_attrs[e, lm] * (LM_TO_L[lm] == l)         (16x4)
//   B[l, c]  = tp_weights[e, l, c] * node_feats[sender, c]    (4x96)
//   messages = A x B  -> six V_WMMA_F32_16X16X4_F32 tiles (c in steps of 16)
// Scatter with global_atomic_add_f32 into out[recv, lm, c]; output (61 MB)
// is L2-resident on MI455X (192 MB L2) so atomics stay on-chip.
//
// A-fragment layout (ISA 7.12.2, 32-bit A 16x4): lane m=lane&15 holds row M=m;
// lanes 0-15: VGPR0=K0, VGPR1=K1; lanes 16-31: VGPR0=K2, VGPR1=K3.
// B-fragment mirrors over N=lane&15 with the same K split.
// C layout: VGPR r holds M=r (lanes 0-15) / M=r+8 (lanes 16-31), N=lane&15.
// ---------------------------------------------------------------------------
__global__ __launch_bounds__(256) void imp_tp_wmma_kernel(
    const float* __restrict__ node_feats,   // [nnodes, 96]
    const float* __restrict__ edge_attrs,   // [nedges, 16]
    const float* __restrict__ tp_weights,   // [nedges, 4, 96]
    const long long* __restrict__ sender,   // [nedges]
    const long long* __restrict__ recv,     // [nedges]
    float* __restrict__ out,                // [nnodes, 16, 96]
    int nedges)
{
  const int lane = threadIdx.x & 31;
  const int wave = threadIdx.x >> 5;
  const long long e = (long long)blockIdx.x * 8 + wave;
  if (e >= nedges) return;                  // wave-uniform: EXEC stays full

  const int n  = lane & 15;                 // row (lm) for A, col (c%16) for B/C
  const int hi = lane >> 4;                 // 0: K={0,1}, 1: K={2,3}
  const int k0 = hi << 1;

  const long long snd = sender[e];
  const long long rcv = recv[e];

  // A fragment: LM_TO_L = [0, 1,1,1, 2,2,2,2,2, 3,3,3,3,3,3,3]
  const float ea     = edge_attrs[e * 16 + n];
  const int   l_of_m = (n == 0) ? 0 : (n < 4) ? 1 : (n < 9) ? 2 : 3;
  v2f a;
  a.x = (l_of_m == k0)     ? ea : 0.0f;
  a.y = (l_of_m == k0 + 1) ? ea : 0.0f;

  const float* wbase = tp_weights + e * 384;      // 4 rows x 96
  const float* sbase = node_feats + snd * 96;
  float*       obase = out + rcv * 1536;          // 16 rows x 96

  // Prefetch edge e+8's 1.5 KB weight block (speculative; OOB is dropped).
  __builtin_prefetch(wbase + 8 * 384 + lane * 4, 0, 1);

#pragma unroll
  for (int t = 0; t < 6; ++t) {
    const int c  = t * 16 + n;
    const float sf = sbase[c];
    v2f b;
    b.x = wbase[ k0      * 96 + c] * sf;
    b.y = wbase[(k0 + 1) * 96 + c] * sf;

    v8f acc = {};
    // (neg_a, A, neg_b, B, c_mod, C, reuse_a, reuse_b)
    acc = __builtin_amdgcn_wmma_f32_16x16x4_f32(
        false, a, false, b, (short)0, acc, false, false);

    // acc[r] = messages[lm = r + 8*hi, c]; each atomic op covers two
    // contiguous 64B rows across the wave.
    float* orow = obase + (8 * hi) * 96 + c;
#pragma unroll
    for (int r = 0; r < 8; ++r) {
      atomicAdd(orow + r * 96, acc[r]);
    }
  }
}

// ---------------------------------------------------------------------------
extern "C" void kernel_launch(void* const* d_in, const int* in_sizes, int n_in,
                              void* d_out, int out_size, void* d_ws, size_t ws_size,
                              hipStream_t stream) {
  const float*     node_feats = (const float*)d_in[0];
  const float*     edge_attrs = (const float*)d_in[1];
  const float*     tp_weights = (const float*)d_in[2];
  const long long* sender     = (const long long*)d_in[3];
  const long long* recv       = (const long long*)d_in[4];
  float*           out        = (float*)d_out;

  const int nedges = in_sizes[3];

  // 1) zero the output
  int n4 = (out_size + 3) / 4;
  imp_zero_kernel<<<(n4 + 255) / 256, 256, 0, stream>>>(out, out_size);

  // 2) one wave per edge, 8 edges per 256-thread block
  int nblocks = (nedges + 7) / 8;
  imp_tp_wmma_kernel<<<nblocks, 256, 0, stream>>>(
      node_feats, edge_attrs, tp_weights, sender, recv, out, nedges);
}